// Affine_87007447482684
// MI455X (gfx1250) — compile-verified
//
#include <hip/hip_runtime.h>
#include <hip/hip_bf16.h>

typedef float v2f __attribute__((ext_vector_type(2)));
typedef float v8f __attribute__((ext_vector_type(8)));

// ---------------------------------------------------------------------------
// Problem constants (from reference setup_inputs): B=64, C=3, H=W=512
// ---------------------------------------------------------------------------
#define BATCH 64
#define CHANS 3
#define IMH   512
#define IMW   512
#define HW    (IMH * IMW)          // 262144 = 1<<18
#define BLK_PER_IMG (HW / 256)     // 1024

// ---------------------------------------------------------------------------
// Kernel 1: theta MLP via full-precision V_WMMA_F32_16X16X4_F32.
// One wave32 block per 16-row batch tile (4 blocks total).
//
// ISA layouts used (CDNA5 ISA §7.12.2, 32-bit matrices):
//   A (16x4, MxK):  lane l<16 -> row M=l,   {v0,v1} = A[M][k0+0], A[M][k0+1]
//                   lane l>=16 -> row M=l-16,{v0,v1} = A[M][k0+2], A[M][k0+3]
//   B (4x16, KxN):  symmetric: lane -> N, low half-wave holds K=k0+0/+1,
//                   high half-wave holds K=k0+2/+3
//   C/D (16x16):    lane n<16: vgpr r -> D[M=r][N=n]; lane 16+n: D[M=8+r][N=n]
// K is padded (6->8) with zeros; N padded to 16 with zero weight columns.
// Padding is done by clamped loads + select-to-zero (no exec divergence).
// ---------------------------------------------------------------------------
__global__ __launch_bounds__(32)
void stn_theta_wmma(const float* __restrict__ noise,   // [64,6]
                    const int*   __restrict__ label,   // [64]
                    const float* __restrict__ W1,      // [8,6]
                    const float* __restrict__ b1,      // [8]
                    const float* __restrict__ W2,      // [6,8]
                    const float* __restrict__ b2,      // [6]
                    const float* __restrict__ W3,      // [1,6]
                    const float* __restrict__ b3,      // [1]
                    float2* __restrict__ trig,         // [64] {cos,sin}
                    float*  __restrict__ outLabel)     // d_out + img elems
{
    const int lane    = threadIdx.x;       // 0..31
    const int n       = lane & 15;         // N column / row index
    const bool hi     = lane >= 16;
    const int rowBase = blockIdx.x * 16;   // batch tile start

    __shared__ float h1s[16][8];           // layer-1 activations
    __shared__ float h2s[16][8];           // layer-2 activations (6 used)

    // ---- Layer 1: h1 = relu(noise[16x6] @ W1^T[6x8] + b1), K padded to 8
    auto loadA1 = [&](int k0) -> v2f {
        const float* rp = noise + (size_t)(rowBase + n) * 6;
        int ka = k0 + (hi ? 2 : 0);
        int kb = ka + 1;
        float ax = rp[ka < 6 ? ka : 5];    // clamped load, masked below
        float ay = rp[kb < 6 ? kb : 5];
        v2f a;
        a.x = (ka < 6) ? ax : 0.0f;
        a.y = (kb < 6) ? ay : 0.0f;
        return a;
    };
    auto loadB1 = [&](int k0) -> v2f {     // B[k][n] = W1[n][k], n<8, k<6
        int ka = k0 + (hi ? 2 : 0);
        int kb = ka + 1;
        const float* wp = W1 + (n < 8 ? n : 7) * 6;
        float bx = wp[ka < 6 ? ka : 5];
        float by = wp[kb < 6 ? kb : 5];
        v2f b;
        b.x = (n < 8 && ka < 6) ? bx : 0.0f;
        b.y = (n < 8 && kb < 6) ? by : 0.0f;
        return b;
    };

    v8f acc1 = {0.f,0.f,0.f,0.f,0.f,0.f,0.f,0.f};
    acc1 = __builtin_amdgcn_wmma_f32_16x16x4_f32(false, loadA1(0), false, loadB1(0),
                                                 (short)0, acc1, false, false);
    acc1 = __builtin_amdgcn_wmma_f32_16x16x4_f32(false, loadA1(4), false, loadB1(4),
                                                 (short)0, acc1, false, false);

    if (n < 8) {
        float bias = b1[n];
        int mBase = hi ? 8 : 0;
        #pragma unroll
        for (int r = 0; r < 8; ++r) {
            float v = acc1[r] + bias;
            h1s[mBase + r][n] = v > 0.0f ? v : 0.0f;
        }
    }
    __syncthreads();

    // ---- Layer 2: h2 = relu(h1[16x8] @ W2^T[8x6] + b2), K=8 exact
    auto loadA2 = [&](int k0) -> v2f {
        int ka = k0 + (hi ? 2 : 0);
        v2f a;
        a.x = h1s[n][ka];
        a.y = h1s[n][ka + 1];
        return a;
    };
    auto loadB2 = [&](int k0) -> v2f {     // B[k][n] = W2[n][k], n<6
        int ka = k0 + (hi ? 2 : 0);
        const float* wp = W2 + (n < 6 ? n : 5) * 8;
        float bx = wp[ka];
        float by = wp[ka + 1];
        v2f b;
        b.x = (n < 6) ? bx : 0.0f;
        b.y = (n < 6) ? by : 0.0f;
        return b;
    };

    v8f acc2 = {0.f,0.f,0.f,0.f,0.f,0.f,0.f,0.f};
    acc2 = __builtin_amdgcn_wmma_f32_16x16x4_f32(false, loadA2(0), false, loadB2(0),
                                                 (short)0, acc2, false, false);
    acc2 = __builtin_amdgcn_wmma_f32_16x16x4_f32(false, loadA2(4), false, loadB2(4),
                                                 (short)0, acc2, false, false);

    if (n < 6) {
        float bias = b2[n];
        int mBase = hi ? 8 : 0;
        #pragma unroll
        for (int r = 0; r < 8; ++r) {
            float v = acc2[r] + bias;
            h2s[mBase + r][n] = v > 0.0f ? v : 0.0f;
        }
    }
    __syncthreads();

    // ---- Layer 3: theta = h2[16x6] @ W3^T[6x1] + b3 (scalar dot per row)
    if (lane < 16) {
        float t = b3[0];
        #pragma unroll
        for (int k = 0; k < 6; ++k)
            t = fmaf(h2s[lane][k], W3[k], t);
        int b = rowBase + lane;
        trig[b] = make_float2(cosf(t), sinf(t));
        outLabel[b] = (float)label[b];
    }
}

// ---------------------------------------------------------------------------
// Kernel 2: rotated bilinear resample (the bandwidth-bound part).
// One thread per (b, y, x); tap addresses/weights reused across 3 channels.
// b is derived purely from blockIdx (HW/256 = 1024 blocks per image), so the
// trig load and both image/output base addresses are wave-uniform (SGPR).
// Zero-padding folded into the bilinear weights (matches reference's
// value*valid*weight since inputs are finite).
// ---------------------------------------------------------------------------
__global__ __launch_bounds__(256)
void stn_sample(const float*  __restrict__ img,   // [64,3,512,512]
                const float2* __restrict__ trig,  // [64] {cos,sin}
                float*        __restrict__ out)   // [64,3,512,512]
{
    const int b    = blockIdx.x >> 10;                           // uniform
    const int rem  = ((blockIdx.x & (BLK_PER_IMG - 1)) << 8) | threadIdx.x;
    const int y    = rem >> 9;
    const int x    = rem & (IMW - 1);

    float2 cs = trig[b];                          // uniform address -> scalarizable
    const float step = 2.0f / (float)(IMW - 1);   // align_corners=True linspace
    float gx = fmaf((float)x, step, -1.0f);
    float gy = fmaf((float)y, step, -1.0f);

    // theta_mat = [[c, -s, 0], [s, c, 0]]
    float sx = cs.x * gx - cs.y * gy;
    float sy = cs.y * gx + cs.x * gy;

    float ix = (sx + 1.0f) * (0.5f * (float)(IMW - 1));
    float iy = (sy + 1.0f) * (0.5f * (float)(IMH - 1));

    float ix0f = floorf(ix), iy0f = floorf(iy);
    int ix0 = (int)ix0f, iy0 = (int)iy0f;
    int ix1 = ix0 + 1,   iy1 = iy0 + 1;

    float wx1 = ix - ix0f, wx0 = 1.0f - wx1;
    float wy1 = iy - iy0f, wy0 = 1.0f - wy1;

    bool vx0 = (ix0 >= 0) && (ix0 < IMW);
    bool vx1 = (ix1 >= 0) && (ix1 < IMW);
    bool vy0 = (iy0 >= 0) && (iy0 < IMH);
    bool vy1 = (iy1 >= 0) && (iy1 < IMH);

    int xc0 = min(max(ix0, 0), IMW - 1);
    int xc1 = min(max(ix1, 0), IMW - 1);
    int yc0 = min(max(iy0, 0), IMH - 1);
    int yc1 = min(max(iy1, 0), IMH - 1);

    float t00 = (vy0 && vx0) ? wy0 * wx0 : 0.0f;
    float t01 = (vy0 && vx1) ? wy0 * wx1 : 0.0f;
    float t10 = (vy1 && vx0) ? wy1 * wx0 : 0.0f;
    float t11 = (vy1 && vx1) ? wy1 * wx1 : 0.0f;

    int off00 = yc0 * IMW + xc0;
    int off01 = yc0 * IMW + xc1;
    int off10 = yc1 * IMW + xc0;
    int off11 = yc1 * IMW + xc1;

    const float* base = img + (size_t)b * (CHANS * HW);   // SGPR base
    float*       ob   = out + (size_t)b * (CHANS * HW) + rem;

    #pragma unroll
    for (int c = 0; c < CHANS; ++c) {
        const float* pc = base + c * HW;
        float v = pc[off00] * t00 + pc[off01] * t01
                + pc[off10] * t10 + pc[off11] * t11;
        __builtin_nontemporal_store(v, ob + c * HW);   // write-once output
    }
}

// ---------------------------------------------------------------------------
extern "C" void kernel_launch(void* const* d_in, const int* in_sizes, int n_in,
                              void* d_out, int out_size, void* d_ws, size_t ws_size,
                              hipStream_t stream) {
    const float* image = (const float*)d_in[0];
    const int*   label = (const int*)  d_in[1];
    const float* noise = (const float*)d_in[2];
    const float* W1    = (const float*)d_in[3];
    const float* b1    = (const float*)d_in[4];
    const float* W2    = (const float*)d_in[5];
    const float* b2    = (const float*)d_in[6];
    const float* W3    = (const float*)d_in[7];
    const float* b3    = (const float*)d_in[8];

    float*  out  = (float*)d_out;
    float2* trig = (float2*)d_ws;                       // 64 * 8B scratch

    const size_t imgElems = (size_t)BATCH * CHANS * HW; // 50,331,648

    stn_theta_wmma<<<BATCH / 16, 32, 0, stream>>>(
        noise, label, W1, b1, W2, b2, W3, b3, trig, out + imgElems);

    stn_sample<<<(BATCH * HW) / 256, 256, 0, stream>>>(image, trig, out);
}